// GAT_27204322853583
// MI455X (gfx1250) — compile-verified
//
#include <hip/hip_runtime.h>
#include <math.h>

typedef __attribute__((ext_vector_type(2))) float v2f;
typedef __attribute__((ext_vector_type(8))) float v8f;

#define FC 128       // feature dim C
#define EDIM 16      // edge feature dim
#define NEG_SLOPE 0.2f

// ---------- helpers ----------
static __device__ __forceinline__ unsigned f2o(float f) {
  unsigned u = __float_as_uint(f);
  return (u & 0x80000000u) ? ~u : (u | 0x80000000u);   // order-preserving map
}
static __device__ __forceinline__ float o2f(unsigned u) {
  return (u & 0x80000000u) ? __uint_as_float(u & 0x7fffffffu) : __uint_as_float(~u);
}
static __device__ __forceinline__ float silu_f(float v) {
  return v / (1.0f + __expf(-v));
}

// ---------- zero init ----------
__global__ void k_zero(float* __restrict__ p, size_t n) {
  size_t i = (size_t)blockIdx.x * blockDim.x + threadIdx.x;
  if (i < n) p[i] = 0.0f;
}

// ---------- edge_attr per-feature mean (partial sums, atomic finish) ----------
__global__ void k_ea_mean(const float* __restrict__ ea, float* __restrict__ acc, int E) {
  float loc[EDIM];
#pragma unroll
  for (int j = 0; j < EDIM; ++j) loc[j] = 0.0f;
  for (int e = blockIdx.x * blockDim.x + threadIdx.x; e < E; e += gridDim.x * blockDim.x) {
    const float4* r = (const float4*)(ea + (size_t)e * EDIM);
#pragma unroll
    for (int j = 0; j < 4; ++j) {
      float4 v = r[j];
      loc[4 * j + 0] += v.x; loc[4 * j + 1] += v.y;
      loc[4 * j + 2] += v.z; loc[4 * j + 3] += v.w;
    }
  }
#pragma unroll
  for (int j = 0; j < EDIM; ++j) atomicAdd(acc + j, loc[j]);
}

// ---------- wvec_l = We_l @ a_e_l  (16 dots of length 128); sloop_l = mean_ea . wvec_l ----------
__global__ void k_prep(const float* __restrict__ We1, const float* __restrict__ ae1,
                       const float* __restrict__ We2, const float* __restrict__ ae2,
                       const float* __restrict__ meanacc,
                       float* __restrict__ wv1, float* __restrict__ wv2,
                       float* __restrict__ sloop, float invE) {
  int t = threadIdx.x;
  if (t < EDIM) {
    float s = 0.0f;
    for (int c = 0; c < FC; ++c) s += We1[t * FC + c] * ae1[c];
    wv1[t] = s;
  } else if (t < 2 * EDIM) {
    int j = t - EDIM; float s = 0.0f;
    for (int c = 0; c < FC; ++c) s += We2[j * FC + c] * ae2[c];
    wv2[j] = s;
  }
  __syncthreads();
  if (t == 0) {
    float s1 = 0.0f, s2 = 0.0f;
    for (int j = 0; j < EDIM; ++j) {
      float m = meanacc[j] * invE;
      s1 += m * wv1[j]; s2 += m * wv2[j];
    }
    sloop[0] = s1; sloop[1] = s2;
  }
}

// ---------- per-edge s_e for both layers in one pass over edge_attr ----------
__global__ void k_edge_se(const float* __restrict__ ea,
                          const float* __restrict__ wv1, const float* __restrict__ wv2,
                          float* __restrict__ sE1, float* __restrict__ sE2, int E) {
  int e = blockIdx.x * blockDim.x + threadIdx.x;
  if (e >= E) return;
  const float4* r = (const float4*)(ea + (size_t)e * EDIM);
  const float4* w1 = (const float4*)wv1;
  const float4* w2 = (const float4*)wv2;
  float s1 = 0.0f, s2 = 0.0f;
#pragma unroll
  for (int j = 0; j < 4; ++j) {
    float4 v = r[j], a = w1[j], b = w2[j];
    s1 += v.x * a.x + v.y * a.y + v.z * a.z + v.w * a.w;
    s2 += v.x * b.x + v.y * b.y + v.z * b.z + v.w * b.w;
  }
  sE1[e] = s1; sE2[e] = s2;
}

// ---------- WMMA fp32 GEMM: H[M,128] = X[M,128] @ W[128,128] ----------
// block: 256 threads = 8 waves; block covers 16 rows x 128 cols; wave w -> cols [16w,16w+16)
// K loop: 32 x V_WMMA_F32_16X16X4_F32
__global__ __launch_bounds__(256) void k_gemm_wmma(const float* __restrict__ X,
                                                   const float* __restrict__ W,
                                                   float* __restrict__ H, int nrows) {
  const int wave = threadIdx.x >> 5;
  const int lane = threadIdx.x & 31;
  const int laneHalf = lane >> 4;     // 0: K lo pair, 1: K hi pair
  const int l15 = lane & 15;
  const int m0 = blockIdx.x * 16;
  const int n0 = wave * 16;

  int rowA = m0 + l15;
  if (rowA >= nrows) rowA = nrows - 1;           // clamp (no EXEC divergence)
  const float* __restrict__ xrow = X + (size_t)rowA * FC;

  v8f acc = {};
#pragma unroll 4
  for (int k = 0; k < FC; k += 4) {
    const int ka = k + laneHalf * 2;
    v2f a, b;
    a.x = xrow[ka];
    a.y = xrow[ka + 1];
    b.x = W[(size_t)ka * FC + n0 + l15];
    b.y = W[(size_t)(ka + 1) * FC + n0 + l15];
    acc = __builtin_amdgcn_wmma_f32_16x16x4_f32(
        /*neg_a=*/false, a, /*neg_b=*/false, b,
        /*c_mod=*/(short)0, acc, /*reuse_a=*/false, /*reuse_b=*/false);
  }

  const int colD = n0 + l15;
  const int rbase = m0 + laneHalf * 8;
  float* __restrict__ hp = H + (size_t)rbase * FC + colD;
  if (m0 + 16 <= nrows) {
    // fast path: full tile, unguarded coalesced stores (uniform scalar branch)
#pragma unroll
    for (int i = 0; i < 8; ++i) hp[(size_t)i * FC] = acc[i];
  } else {
#pragma unroll
    for (int i = 0; i < 8; ++i) {
      if (rbase + i < nrows) hp[(size_t)i * FC] = acc[i];
    }
  }
}

// ---------- per-node scores: s_src = h . a_src, s_dst = h . a_dst (wave per node) ----------
__global__ void k_node_scores(const float* __restrict__ H,
                              const float* __restrict__ asrc, const float* __restrict__ adst,
                              float* __restrict__ ssrc, float* __restrict__ sdst, int n) {
  int wid = (int)(((size_t)blockIdx.x * blockDim.x + threadIdx.x) >> 5);
  int lane = threadIdx.x & 31;
  if (wid >= n) return;
  float4 v = ((const float4*)(H + (size_t)wid * FC))[lane];
  float4 a = ((const float4*)asrc)[lane];
  float4 d = ((const float4*)adst)[lane];
  float s1 = v.x * a.x + v.y * a.y + v.z * a.z + v.w * a.w;
  float s2 = v.x * d.x + v.y * d.y + v.z * d.z + v.w * d.w;
#pragma unroll
  for (int off = 16; off > 0; off >>= 1) {
    s1 += __shfl_xor(s1, off, 32);
    s2 += __shfl_xor(s2, off, 32);
  }
  if (lane == 0) { ssrc[wid] = s1; sdst[wid] = s2; }
}

// ---------- edge pass 1: alpha = leaky_relu(...); segment max via ordered-uint atomicMax ----------
__global__ void k_edge_alpha_max(const int* __restrict__ ei, const float* __restrict__ sE,
                                 const float* __restrict__ sloop,
                                 const float* __restrict__ ssrc, const float* __restrict__ sdst,
                                 float* __restrict__ alpha, unsigned* __restrict__ amax,
                                 int E, int N) {
  int e = blockIdx.x * blockDim.x + threadIdx.x;
  if (e >= E + N) return;
  int s, d; float se;
  if (e < E) { s = ei[e]; d = ei[E + e]; se = sE[e]; }
  else       { s = d = e - E; se = sloop[0]; }
  float a = ssrc[s] + sdst[d] + se;
  a = a > 0.0f ? a : NEG_SLOPE * a;
  alpha[e] = a;
  atomicMax(amax + d, f2o(a));
}

// ---------- edge pass 2: ex = exp(alpha - max); segment sum ----------
__global__ void k_edge_expsum(const int* __restrict__ ei, float* __restrict__ alpha,
                              const unsigned* __restrict__ amax, float* __restrict__ den,
                              int E, int N) {
  int e = blockIdx.x * blockDim.x + threadIdx.x;
  if (e >= E + N) return;
  int d = (e < E) ? ei[E + e] : (e - E);
  float v = __expf(alpha[e] - o2f(amax[d]));
  alpha[e] = v;
  atomicAdd(den + d, v);
}

// ---------- edge pass 3: agg[dst] += (ex/den[dst]) * h[src]  (wave per edge, float4 rows) ----------
__global__ void k_edge_scatter(const int* __restrict__ ei, const float* __restrict__ ex,
                               const float* __restrict__ den, const float* __restrict__ H,
                               float* __restrict__ agg, int E, int N) {
  int wid = (int)(((size_t)blockIdx.x * blockDim.x + threadIdx.x) >> 5);
  int lane = threadIdx.x & 31;
  if (wid >= E + N) return;
  int s, d;
  if (wid < E) { s = ei[wid]; d = ei[E + wid]; }
  else         { s = d = wid - E; }
  float w = ex[wid] / den[d];
  float4 v = ((const float4*)(H + (size_t)s * FC))[lane];
  float* o = agg + (size_t)d * FC + lane * 4;
  atomicAdd(o + 0, w * v.x);
  atomicAdd(o + 1, w * v.y);
  atomicAdd(o + 2, w * v.z);
  atomicAdd(o + 3, w * v.w);
}

// ---------- epilogue: y = silu(agg + bias) + skip ----------
__global__ void k_epilogue(const float* __restrict__ agg, const float* __restrict__ bias,
                           const float* __restrict__ skip, float* __restrict__ y, size_t n) {
  size_t i = (size_t)blockIdx.x * blockDim.x + threadIdx.x;
  if (i >= n) return;
  float v = agg[i] + bias[i & (FC - 1)];
  y[i] = silu_f(v) + skip[i];
}

// ---------- LayerNorm + global mean pool accumulate (wave per node) ----------
__global__ void k_ln_pool(const float* __restrict__ Hf, const int* __restrict__ batch,
                          const float* __restrict__ g, const float* __restrict__ b,
                          float* __restrict__ pool, float* __restrict__ cnt, int n) {
  int wid = (int)(((size_t)blockIdx.x * blockDim.x + threadIdx.x) >> 5);
  int lane = threadIdx.x & 31;
  if (wid >= n) return;
  float4 v = ((const float4*)(Hf + (size_t)wid * FC))[lane];
  float s  = v.x + v.y + v.z + v.w;
  float sq = v.x * v.x + v.y * v.y + v.z * v.z + v.w * v.w;
#pragma unroll
  for (int off = 16; off > 0; off >>= 1) {
    s  += __shfl_xor(s, off, 32);
    sq += __shfl_xor(sq, off, 32);
  }
  float mu = s * (1.0f / FC);
  float var = sq * (1.0f / FC) - mu * mu;
  float rs = rsqrtf(var + 1e-5f);
  float4 gg = ((const float4*)g)[lane];
  float4 bb = ((const float4*)b)[lane];
  float4 o;
  o.x = (v.x - mu) * rs * gg.x + bb.x;
  o.y = (v.y - mu) * rs * gg.y + bb.y;
  o.z = (v.z - mu) * rs * gg.z + bb.z;
  o.w = (v.w - mu) * rs * gg.w + bb.w;
  int grp = batch[wid];
  float* p = pool + (size_t)grp * FC + lane * 4;
  atomicAdd(p + 0, o.x);
  atomicAdd(p + 1, o.y);
  atomicAdd(p + 2, o.z);
  atomicAdd(p + 3, o.w);
  if (lane == 0) atomicAdd(cnt + grp, 1.0f);
}

// ---------- final divide ----------
__global__ void k_pool_div(const float* __restrict__ pool, const float* __restrict__ cnt,
                           float* __restrict__ out, int total) {
  int i = blockIdx.x * blockDim.x + threadIdx.x;
  if (i >= total) return;
  float c = cnt[i >> 7];
  out[i] = pool[i] / fmaxf(c, 1.0f);
}

// ==========================================================================
static inline int cdiv(size_t a, size_t b) { return (int)((a + b - 1) / b); }

static void launch_layer(const float* xin, const float* W, const float* asrc,
                         const float* adst, const float* bias, const float* sE,
                         const float* sloop, const float* skip, float* yout,
                         float* H, float* AGG, float* SSRC, float* SDST,
                         unsigned* AMAX, float* DEN, float* EX,
                         const int* ei, int N, int E, hipStream_t stream) {
  const size_t NC = (size_t)N * FC;
  const int TOT = E + N;
  // h = x @ W   (WMMA)
  k_gemm_wmma<<<cdiv(N, 16), 256, 0, stream>>>(xin, W, H, N);
  // node scores
  k_node_scores<<<cdiv((size_t)N * 32, 256), 256, 0, stream>>>(H, asrc, adst, SSRC, SDST, N);
  // reset accumulators
  k_zero<<<cdiv(NC, 256), 256, 0, stream>>>(AGG, NC);
  k_zero<<<cdiv((size_t)N, 256), 256, 0, stream>>>((float*)AMAX, (size_t)N);
  k_zero<<<cdiv((size_t)N, 256), 256, 0, stream>>>(DEN, (size_t)N);
  // softmax over incoming edges
  k_edge_alpha_max<<<cdiv((size_t)TOT, 256), 256, 0, stream>>>(ei, sE, sloop, SSRC, SDST, EX, AMAX, E, N);
  k_edge_expsum<<<cdiv((size_t)TOT, 256), 256, 0, stream>>>(ei, EX, AMAX, DEN, E, N);
  // weighted aggregation
  k_edge_scatter<<<cdiv((size_t)TOT * 32, 256), 256, 0, stream>>>(ei, EX, DEN, H, AGG, E, N);
  // y = silu(agg + b) + skip
  k_epilogue<<<cdiv(NC, 256), 256, 0, stream>>>(AGG, bias, skip, yout, NC);
}

extern "C" void kernel_launch(void* const* d_in, const int* in_sizes, int n_in,
                              void* d_out, int out_size, void* d_ws, size_t ws_size,
                              hipStream_t stream) {
  const float* x   = (const float*)d_in[0];
  const int*   ei  = (const int*)d_in[1];
  const float* ea  = (const float*)d_in[2];
  const int*   bat = (const int*)d_in[3];
  const float* W1  = (const float*)d_in[4];
  const float* as1 = (const float*)d_in[5];
  const float* ad1 = (const float*)d_in[6];
  const float* We1 = (const float*)d_in[7];
  const float* ae1 = (const float*)d_in[8];
  const float* b1  = (const float*)d_in[9];
  const float* W2  = (const float*)d_in[10];
  const float* as2 = (const float*)d_in[11];
  const float* ad2 = (const float*)d_in[12];
  const float* We2 = (const float*)d_in[13];
  const float* ae2 = (const float*)d_in[14];
  const float* b2  = (const float*)d_in[15];
  const float* lng = (const float*)d_in[16];
  const float* lnb = (const float*)d_in[17];
  float* out = (float*)d_out;

  const int N = in_sizes[0] / FC;
  const int E = in_sizes[1] / 2;
  const int G = out_size / FC;
  const size_t NC = (size_t)N * FC;

  // workspace layout (all fp32 slots; big arrays first, chunks packed back to back, CNT padded to 16 floats)
  float* H    = (float*)d_ws;
  float* AGG  = H + NC;
  float* Y1   = AGG + NC;
  float* SE1  = Y1 + NC;
  float* SE2  = SE1 + E;
  float* EX   = SE2 + E;
  float* SSRC = EX + ((size_t)E + N);
  float* SDST = SSRC + N;
  unsigned* AMAX = (unsigned*)(SDST + N);
  float* DEN  = (float*)(AMAX + N);
  float* POOL = DEN + N;
  float* CNT  = POOL + (size_t)G * FC;
  float* MEAN = CNT + (((size_t)G + 15) & ~(size_t)15);
  float* WV1  = MEAN + EDIM;
  float* WV2  = WV1 + EDIM;
  float* SLOOP = WV2 + EDIM;   // [0]=layer1, [1]=layer2

  // ---- precompute: edge mean, wvec = We@a_e, s_e per edge (both layers, one ea pass)
  k_zero<<<1, 64, 0, stream>>>(MEAN, (size_t)(3 * EDIM + 2));
  k_ea_mean<<<64, 256, 0, stream>>>(ea, MEAN, E);
  k_prep<<<1, 64, 0, stream>>>(We1, ae1, We2, ae2, MEAN, WV1, WV2, SLOOP, 1.0f / (float)E);
  k_edge_se<<<cdiv((size_t)E, 256), 256, 0, stream>>>(ea, WV1, WV2, SE1, SE2, E);

  // ---- layer 1: skip = x, out -> Y1
  launch_layer(x, W1, as1, ad1, b1, SE1, SLOOP + 0, x, Y1,
               H, AGG, SSRC, SDST, AMAX, DEN, EX, ei, N, E, stream);
  // ---- layer 2: skip = Y1, out -> AGG (in-place epilogue)
  launch_layer(Y1, W2, as2, ad2, b2, SE2, SLOOP + 1, Y1, AGG,
               H, AGG, SSRC, SDST, AMAX, DEN, EX, ei, N, E, stream);

  // ---- LayerNorm + global mean pool
  k_zero<<<cdiv((size_t)G * FC + G, 256), 256, 0, stream>>>(POOL, (size_t)G * FC + G); // POOL and CNT contiguous
  k_ln_pool<<<cdiv((size_t)N * 32, 256), 256, 0, stream>>>(AGG, bat, lng, lnb, POOL, CNT, N);
  k_pool_div<<<cdiv((size_t)G * FC, 256), 256, 0, stream>>>(POOL, CNT, out, G * FC);
}